// RQBottleneck_28415503631024
// MI455X (gfx1250) — compile-verified
//
#include <hip/hip_runtime.h>

typedef __attribute__((ext_vector_type(2))) float v2f;
typedef __attribute__((ext_vector_type(8))) float v8f;

#define T_TOKENS   65536
#define E_DIM      64
#define N_CODES    2048
#define N_LVL      4
#define RSTR       68          // LDS row stride (floats), padded vs 64 for bank-conflict-free frags
#define Z_SIZE     4194304     // 64*64*64*16
#define NBLK       4096

// ---------------------------------------------------------------------------
// Kernel 1: half squared norms of all codebook rows: cnorm[lvl*2048+c] = 0.5*||c||^2
// ---------------------------------------------------------------------------
__global__ __launch_bounds__(128) void rq_cnorm_kernel(const float* __restrict__ cb,
                                                       float* __restrict__ cnorm) {
  int row = blockIdx.x * 128 + threadIdx.x;              // 0..8191
  const float4* p = (const float4*)(cb + (size_t)row * E_DIM);
  float s = 0.f;
#pragma unroll
  for (int i = 0; i < 16; ++i) {
    float4 v = p[i];
    s += v.x * v.x + v.y * v.y + v.z * v.z + v.w * v.w;
  }
  cnorm[row] = 0.5f * s;
}

// ---------------------------------------------------------------------------
// Kernel 2: main RQ kernel. One WG = 16 tokens through all 4 levels.
// ---------------------------------------------------------------------------
__global__ __launch_bounds__(128) void rq_main_kernel(const float* __restrict__ x,
                                                      const float* __restrict__ cb,
                                                      const float* __restrict__ cnorm,
                                                      float* __restrict__ lossPart,
                                                      float* __restrict__ out) {
  __shared__ float Rl[16 * RSTR];   // residual tile [16][64] padded
  __shared__ float Ql[16 * RSTR];   // quant tile
  __shared__ float wminV[4 * 16];
  __shared__ int   wminI[4 * 16];
  __shared__ int   bidx[16];
  __shared__ float lpart[4];

  const int tid  = threadIdx.x;
  const int lane = tid & 31;
  const int wave = tid >> 5;
  const int t0   = blockIdx.x * 16;

  const int tl = tid >> 3;          // local token 0..15 (8 threads per token)
  const int eb = (tid & 7) * 8;     // 8 contiguous embed dims per thread

  // ---- load x in code-shape: e = (rh*2+rw)*16 + d ; contiguous runs of 8 in x ----
  {
    int t = t0 + tl;
    int b = t >> 10, hc = (t >> 5) & 31, wc = t & 31;
    int rh = eb >> 5, rw = (eb >> 4) & 1, d0 = eb & 15;
    const float4* src = (const float4*)(x + ((((size_t)b * 64 + hc * 2 + rh) * 64) + (wc * 2 + rw)) * 16 + d0);
    float4 v0 = src[0], v1 = src[1];
    float* rd = &Rl[tl * RSTR + eb];
    rd[0] = v0.x; rd[1] = v0.y; rd[2] = v0.z; rd[3] = v0.w;
    rd[4] = v1.x; rd[5] = v1.y; rd[6] = v1.z; rd[7] = v1.w;
    float* qd = &Ql[tl * RSTR + eb];
#pragma unroll
    for (int k = 0; k < 8; ++k) qd[k] = 0.f;
  }
  __syncthreads();

  const int m16 = lane & 15;
  const int g   = lane >> 4;

  for (int lvl = 0; lvl < N_LVL; ++lvl) {
    const float* cbL = cb + (size_t)lvl * N_CODES * E_DIM;
    const float* cnL = cnorm + lvl * N_CODES;

    // ---- preload A fragments (16x4 f32 layout: lane<16 K=4s+{0,1}; lane>=16 K=4s+{2,3}) ----
    v2f a[16];
    {
      const float* rp = &Rl[m16 * RSTR + 2 * g];
#pragma unroll
      for (int s = 0; s < 16; ++s) { a[s].x = rp[4 * s]; a[s].y = rp[4 * s + 1]; }
    }

    float bv[8];
    int   bi[8];
#pragma unroll
    for (int r = 0; r < 8; ++r) { bv[r] = 3.4e38f; bi[r] = 0; }

    // ---- double-buffered B-tile sweep: 32 tiles of N=16 per wave ------------
    // loadTile: independent 16-load clause; compute: 16-deep WMMA accum chain.
    auto loadTile = [&](v2f (&bf)[16], float& cn, int j) {
      int n = (wave * 32 + j) * 16 + m16;                 // this lane's code row
      const v2f* brow = (const v2f*)(cbL + (size_t)n * E_DIM);
#pragma unroll
      for (int s = 0; s < 16; ++s) bf[s] = brow[2 * s + g];
      cn = cnL[n];
    };
    auto computeTile = [&](const v2f (&bf)[16], float cn, int j) {
      v8f acc = {};
#pragma unroll
      for (int s = 0; s < 16; ++s)
        acc = __builtin_amdgcn_wmma_f32_16x16x4_f32(false, a[s], false, bf[s],
                                                    (short)0, acc, false, false);
      int n = (wave * 32 + j) * 16 + m16;
      // acc[r] : token m = r + 8*g, code n. surrogate dist = 0.5||c||^2 - r.c
      // NOTE: n is strictly increasing over j within a lane, so first-index
      // tie-break == "strictly less than" -> branchless cmp+cndmask only.
#pragma unroll
      for (int r = 0; r < 8; ++r) {
        float v = cn - acc[r];
        bool take = v < bv[r];
        bv[r] = take ? v : bv[r];
        bi[r] = take ? n : bi[r];
      }
    };

    v2f b0[16], b1[16];
    float cn0, cn1;
    loadTile(b0, cn0, 0);
#pragma unroll 1
    for (int j = 0; j < 32; j += 2) {
      loadTile(b1, cn1, j + 1);          // prefetch tile j+1 while computing j
      computeTile(b0, cn0, j);
      if (j + 2 < 32) loadTile(b0, cn0, j + 2);
      computeTile(b1, cn1, j + 1);
    }

    // ---- argmin across the 16 lanes of each half-wave (xor masks < 16 stay in-half) ----
    // Ties across lanes need the first-index rule: branchless bitwise form.
#pragma unroll
    for (int off = 8; off >= 1; off >>= 1) {
#pragma unroll
      for (int r = 0; r < 8; ++r) {
        float ov = __shfl_xor(bv[r], off, 32);
        int   oi = __shfl_xor(bi[r], off, 32);
        int take = (int)(ov < bv[r]) | ((int)(ov == bv[r]) & (int)(oi < bi[r]));
        bv[r] = take ? ov : bv[r];
        bi[r] = take ? oi : bi[r];
      }
    }
    if (m16 == 0) {
#pragma unroll
      for (int r = 0; r < 8; ++r) {
        wminV[wave * 16 + r + 8 * g] = bv[r];
        wminI[wave * 16 + r + 8 * g] = bi[r];
      }
    }
    __syncthreads();

    // ---- combine 4 waves per token; emit code ----
    if (tid < 16) {
      float v = wminV[tid];
      int   i = wminI[tid];
#pragma unroll
      for (int w = 1; w < 4; ++w) {
        float ov = wminV[w * 16 + tid];
        int   oi = wminI[w * 16 + tid];
        int take = (int)(ov < v) | ((int)(ov == v) & (int)(oi < i));
        v = take ? ov : v;
        i = take ? oi : i;
      }
      bidx[tid] = i;
      out[Z_SIZE + 1 + (size_t)(t0 + tid) * N_LVL + lvl] = (float)i;
    }
    __syncthreads();

    // ---- residual/quant update + per-level loss (residual == x_code - quant) ----
    float ls = 0.f;
    {
      const float* e  = cbL + (size_t)bidx[tl] * E_DIM + eb;
      float*       rd = &Rl[tl * RSTR + eb];
      float*       qd = &Ql[tl * RSTR + eb];
#pragma unroll
      for (int k = 0; k < 8; ++k) {
        float ev = e[k];
        float rv = rd[k] - ev;
        rd[k] = rv;
        qd[k] += ev;
        ls += rv * rv;
      }
    }
#pragma unroll
    for (int off = 16; off >= 1; off >>= 1) ls += __shfl_xor(ls, off, 32);
    if (lane == 0) lpart[wave] = ls;
    __syncthreads();
    if (tid == 0)
      lossPart[lvl * NBLK + blockIdx.x] = lpart[0] + lpart[1] + lpart[2] + lpart[3];
    __syncthreads();
  }

  // ---- z_q == quant (STE is value-identity); scatter back to latent shape ----
  {
    int t = t0 + tl;
    int b = t >> 10, hc = (t >> 5) & 31, wc = t & 31;
    int rh = eb >> 5, rw = (eb >> 4) & 1, d0 = eb & 15;
    float* dst = out + ((((size_t)b * 64 + hc * 2 + rh) * 64) + (wc * 2 + rw)) * 16 + d0;
    const float* qd = &Ql[tl * RSTR + eb];
#pragma unroll
    for (int k = 0; k < 8; ++k) dst[k] = qd[k];
  }
}

// ---------------------------------------------------------------------------
// Kernel 3: deterministic loss reduction over 4*4096 per-block partials
// ---------------------------------------------------------------------------
__global__ __launch_bounds__(256) void rq_finalize_kernel(const float* __restrict__ part,
                                                          float* __restrict__ out) {
  __shared__ float sm[256];
  float s = 0.f;
  for (int i = threadIdx.x; i < N_LVL * NBLK; i += 256) s += part[i];
  sm[threadIdx.x] = s;
  __syncthreads();
  for (int off = 128; off > 0; off >>= 1) {
    if (threadIdx.x < off) sm[threadIdx.x] += sm[threadIdx.x + off];
    __syncthreads();
  }
  if (threadIdx.x == 0)
    out[Z_SIZE] = sm[0] / (4.0f * 4194304.0f);   // mean over levels of mean over T*E
}

// ---------------------------------------------------------------------------
extern "C" void kernel_launch(void* const* d_in, const int* in_sizes, int n_in,
                              void* d_out, int out_size, void* d_ws, size_t ws_size,
                              hipStream_t stream) {
  const float* x  = (const float*)d_in[0];   // [64,64,64,16] f32
  const float* cb = (const float*)d_in[1];   // [4,2048,64]  f32
  float* out = (float*)d_out;                // z_q (4194304) | loss (1) | codes (262144)

  float* cnorm    = (float*)d_ws;            // 8192 floats
  float* lossPart = cnorm + N_LVL * N_CODES; // 4*4096 floats

  rq_cnorm_kernel<<<64, 128, 0, stream>>>(cb, cnorm);
  rq_main_kernel<<<NBLK, 128, 0, stream>>>(x, cb, cnorm, lossPart, out);
  rq_finalize_kernel<<<1, 256, 0, stream>>>(lossPart, out);
}